// LSTM_11802570130221
// MI455X (gfx1250) — compile-verified
//
#include <hip/hip_runtime.h>

// LSTM final-state kernel for MI455X (gfx1250, wave32).
// 16 batch rows per wave -> 256 independent recurrence chains (one per SIMD).
// Per step: gates^T(32x16) = [W_hh | W_ih | bias] (32x12) @ [h; x; 1]^T (12x16)
// as two 16x16 tiles, each a chain of three V_WMMA_F32_16X16X4_F32.
// The x/bias WMMA (K=8..11) depends only on prefetched x, so it runs OFF the
// serial h->h dependency chain; only the two W_hh WMMAs are on it.
// Half-wave data movement uses v_permlanex16 (VALU, no LDS round-trip).
// x is double-buffered in two STATICALLY indexed register arrays (xbA/xbB)
// so the buffer cannot be demoted to LDS/scratch.

typedef float v2f __attribute__((ext_vector_type(2)));
typedef float v8f __attribute__((ext_vector_type(8)));

#define SEQ   4096
#define NBAT  4096
#define NIN   2
#define HID   8

__device__ __forceinline__ float ftanh(float x) {
#if __has_builtin(__builtin_amdgcn_tanhf)
  return __builtin_amdgcn_tanhf(x);            // v_tanh_f32 (gfx1250 TRANS op)
#elif __has_builtin(__builtin_amdgcn_tanh_f32)
  return __builtin_amdgcn_tanh_f32(x);
#else
  float e = __expf(-2.0f * x);
  return (1.0f - e) / (1.0f + e);
#endif
}

// Swap the two 16-lane halves of a wave32 (lane i <-> lane i^16).
__device__ __forceinline__ float swap16(float v) {
#if __has_builtin(__builtin_amdgcn_permlanex16)
  int r = __builtin_amdgcn_permlanex16(
      __builtin_bit_cast(int, v), __builtin_bit_cast(int, v),
      0x76543210u, 0xFEDCBA98u, false, false);   // identity cross-half selects
  return __builtin_bit_cast(float, r);
#else
  return __shfl_xor(v, 16, 32);
#endif
}

// Prefetch one 8-step block of x for this lane's batch element.
#define LOADBLK(XB, T0)                                                      \
  _Pragma("unroll")                                                          \
  for (int j = 0; j < 8; ++j) XB[j] = xv[(size_t)((T0) + j) * NBAT + bb];

// Run 8 LSTM steps consuming a statically indexed x block.
#define STEP8(XB)                                                            \
  _Pragma("unroll")                                                          \
  for (int j = 0; j < 8; ++j) {                                              \
    const float x0 = XB[j].x;                                                \
    const float x1 = XB[j].y;                                                \
    /* B operand K=8..11: [x0,x1] low half / [1,0] high half (pure selects) */\
    v2f b_x;                                                                 \
    b_x[0] = half ? 1.0f : x0;                                               \
    b_x[1] = half ? 0.0f : x1;                                               \
    /* Off-chain WMMAs: xproj + bias while h is still in flight */           \
    v8f G0 = __builtin_amdgcn_wmma_f32_16x16x4_f32(                          \
        false, aG0_x, false, b_x, (short)0, vzero, false, false);            \
    v8f G1 = __builtin_amdgcn_wmma_f32_16x16x4_f32(                          \
        false, aG1_x, false, b_x, (short)0, vzero, false, false);            \
    /* B operands = h^T, ISA B-layout: vgpr v, lane L -> K = v + 2*(L/16) */ \
    const float h2s = swap16(h[2]);                                          \
    const float h3s = swap16(h[3]);                                          \
    const float h6s = swap16(h[6]);                                          \
    const float h7s = swap16(h[7]);                                          \
    v2f b_lo, b_hi;                                                          \
    b_lo[0] = half ? h2s : h[0];                                             \
    b_lo[1] = half ? h3s : h[1];                                             \
    b_hi[0] = half ? h6s : h[4];                                             \
    b_hi[1] = half ? h7s : h[5];                                             \
    /* On-chain WMMAs: gates += W_hh * h^T (K=8 as two K=4) */               \
    G0 = __builtin_amdgcn_wmma_f32_16x16x4_f32(                              \
        false, aG0_lo, false, b_lo, (short)0, G0, false, false);             \
    G0 = __builtin_amdgcn_wmma_f32_16x16x4_f32(                              \
        false, aG0_hi, false, b_hi, (short)0, G0, false, false);             \
    G1 = __builtin_amdgcn_wmma_f32_16x16x4_f32(                              \
        false, aG1_lo, false, b_lo, (short)0, G1, false, false);             \
    G1 = __builtin_amdgcn_wmma_f32_16x16x4_f32(                              \
        false, aG1_hi, false, b_hi, (short)0, G1, false, false);             \
    /* Activations: G0 -> sigmoid (i|f); G1 -> tanh(g) low / sigmoid(o) hi */\
    _Pragma("unroll")                                                        \
    for (int r = 0; r < 8; ++r)                                              \
      G0[r] = fmaf(ftanh(0.5f * G0[r]), 0.5f, 0.5f);                         \
    _Pragma("unroll")                                                        \
    for (int r = 0; r < 8; ++r)                                              \
      G1[r] = fmaf(ftanh(sG1 * G1[r]), aG1, bG1);                            \
    /* State update in lanes 0..15 (f,o pulled from the high lane-half) */   \
    _Pragma("unroll")                                                        \
    for (int r = 0; r < 8; ++r) {                                            \
      const float fv = swap16(G0[r]);                                        \
      const float ov = swap16(G1[r]);                                        \
      c[r] = fmaf(fv, c[r], G0[r] * G1[r]);   /* c = f*c + i*g   */          \
      h[r] = ov * ftanh(c[r]);                /* h = o*tanh(c)   */          \
    }                                                                        \
  }

__global__ __launch_bounds__(32) void lstm_wmma_kernel(
    const float* __restrict__ x,     // (SEQ, NBAT, NIN)
    const float* __restrict__ h0,    // (1, NBAT, HID)
    const float* __restrict__ c0,    // (1, NBAT, HID)
    const float* __restrict__ W_ih,  // (4H, NIN)
    const float* __restrict__ W_hh,  // (4H, HID)
    const float* __restrict__ b_ih,  // (4H)
    const float* __restrict__ b_hh,  // (4H)
    float* __restrict__ out)         // (1, NBAT, HID)
{
  const int lane = threadIdx.x;      // 0..31
  const int half = lane >> 4;        // lane-half: selects K-pair / M+8 rows
  const int m    = lane & 15;        // A-row index / batch column
  const int base = blockIdx.x * 16;  // batch tile base

  // ---------------- stationary A operands, ISA A-layout:
  // lane L, vgpr v  ->  A[L%16][v + 2*(L/16)]  (K base +0 / +4 / +8)
  v2f aG0_lo, aG0_hi, aG1_lo, aG1_hi;   // W_hh rows (K = 0..7)
  v2f aG0_x, aG1_x;                     // [W_ih | bias | 0] (K = 8..11)
  {
    const int k = 2 * half;
    const float* w0 = W_hh + (size_t)m * HID;          // rows 0..15  (i,f)
    const float* w1 = W_hh + (size_t)(m + 16) * HID;   // rows 16..31 (g,o)
    aG0_lo[0] = w0[k];     aG0_lo[1] = w0[k + 1];
    aG0_hi[0] = w0[k + 4]; aG0_hi[1] = w0[k + 5];
    aG1_lo[0] = w1[k];     aG1_lo[1] = w1[k + 1];
    aG1_hi[0] = w1[k + 4]; aG1_hi[1] = w1[k + 5];
    // K=8: W_ih[:,0], K=9: W_ih[:,1] (low half); K=10: b_ih+b_hh, K=11: 0 (high)
    const float bias0 = b_ih[m] + b_hh[m];
    const float bias1 = b_ih[m + 16] + b_hh[m + 16];
    aG0_x[0] = half ? bias0 : W_ih[m * NIN + 0];
    aG0_x[1] = half ? 0.0f  : W_ih[m * NIN + 1];
    aG1_x[0] = half ? bias1 : W_ih[(m + 16) * NIN + 0];
    aG1_x[1] = half ? 0.0f  : W_ih[(m + 16) * NIN + 1];
  }

  // G1 activation: tanh in low lanes (g gate), sigmoid in high lanes (o gate)
  // via sigmoid(x) = 0.5*tanh(0.5x)+0.5, selected with lane-half constants.
  const float sG1 = half ? 0.5f : 1.0f;
  const float aG1 = half ? 0.5f : 1.0f;
  const float bG1 = half ? 0.5f : 0.0f;

  // ---------------- recurrent state (valid in lanes 0..15, batch = lane)
  float c[8], h[8];
  {
    const float* hp = h0 + (size_t)(base + m) * HID;
    const float* cp = c0 + (size_t)(base + m) * HID;
#pragma unroll
    for (int r = 0; r < 8; ++r) { h[r] = hp[r]; c[r] = cp[r]; }
  }

  // ---------------- x double-buffer: two statically named register blocks
  const int bb = base + m;                           // this lane's batch
  const float2* __restrict__ xv = (const float2*)x;  // x[t][b] as float2
  const v8f vzero = {};                              // zero C-operand

  float2 xbA[8], xbB[8];
  LOADBLK(xbA, 0)

#pragma unroll 1
  for (int blk = 0; blk < SEQ / 8; blk += 2) {
    LOADBLK(xbB, (blk + 1) * 8)          // blk+1 <= 511: always in range
    STEP8(xbA)
    if (blk + 2 < SEQ / 8) {
      LOADBLK(xbA, (blk + 2) * 8)
    }
    STEP8(xbB)
  }

  // ---------------- final h_n (1, NBAT, HID)
  if (lane < 16) {
    float* op = out + (size_t)(base + lane) * HID;
#pragma unroll
    for (int r = 0; r < 8; ++r) op[r] = h[r];
  }
}

extern "C" void kernel_launch(void* const* d_in, const int* in_sizes, int n_in,
                              void* d_out, int out_size, void* d_ws, size_t ws_size,
                              hipStream_t stream) {
  (void)in_sizes; (void)n_in; (void)out_size; (void)d_ws; (void)ws_size;
  const float* x    = (const float*)d_in[0];
  const float* h0   = (const float*)d_in[1];
  const float* c0   = (const float*)d_in[2];
  const float* W_ih = (const float*)d_in[3];
  const float* W_hh = (const float*)d_in[4];
  const float* b_ih = (const float*)d_in[5];
  const float* b_hh = (const float*)d_in[6];
  float* out = (float*)d_out;

  // 256 batch tiles of 16 -> 256 independent single-wave chains.
  lstm_wmma_kernel<<<NBAT / 16, 32, 0, stream>>>(x, h0, c0, W_ih, W_hh,
                                                 b_ih, b_hh, out);
}